// ConvSpikeEncoder_83485574300330
// MI455X (gfx1250) — compile-verified
//
#include <hip/hip_runtime.h>

typedef float v2f __attribute__((ext_vector_type(2)));
typedef float v8f __attribute__((ext_vector_type(8)));

#define B_     64
#define T_     512
#define CIN_   32
#define HID_   128
#define TS_    4
#define COUT_  512
#define KW_    3
#define KK_    96         // CIN_*KW_ : GEMM reduction dim
#define NSTEP  2048       // T_*TS_
#define NCHAIN 8192       // B_*HID_

#define XROWS  130        // 128 tile rows + 1 halo row each side

// ---------------------------------------------------------------------------
// Kernel 1: Conv1d(k=3,pad=1) + bias + BN(eval) as an LDS-staged WMMA GEMM.
//   M = B*T = 32768, N = C_OUT = 512, K = 96.
//   Block = 256 threads = 8 waves. All 8 waves share one 16-wide N panel and
//   cover 8 consecutive 16-row M tiles (128 time rows inside one image).
//   Prologue stages x rows (halo zero-padded) and the weight panel into LDS
//   with wide coalesced loads; the K loop is then branch-free:
//   24 x (2 ds_load_b64 + v_wmma_f32_16x16x4_f32).
// ---------------------------------------------------------------------------
__global__ __launch_bounds__(256) void conv_bn_wmma_kernel(
    const float* __restrict__ x,   const float* __restrict__ w,
    const float* __restrict__ cb,  const float* __restrict__ gma,
    const float* __restrict__ bta, const float* __restrict__ mu,
    const float* __restrict__ var, float* __restrict__ h) {
  __shared__ float ldsX[XROWS * CIN_];   // [row][ci], row = t - t0_block + 1
  __shared__ float ldsB[(KK_ / 2) * 32]; // [kk>>1][col*2 + (kk&1)]

  const int tid  = threadIdx.x;
  const int lane = tid & 31;
  const int wave = tid >> 5;

  const int n_tile   = blockIdx.x & 31;        // 32 N tiles (512/16)
  const int m_block  = blockIdx.x >> 5;        // 256 row-blocks of 128 rows
  const int n0       = n_tile << 4;
  const int m0_block = m_block << 7;           // 128 rows per block
  const int b        = m0_block >> 9;          // image index (T_=512)
  const int t0_block = m0_block & (T_ - 1);

  const float* xb = x + (size_t)b * (T_ * CIN_);

  // ---- Stage weight panel: ldsB[(kk>>1)*32 + col*2 + (kk&1)] = W'[kk][n0+col]
  //      where W'[kk][n] = conv_w[n, kk&31, kk>>5].
  for (int i = tid; i < KK_ * 16; i += 256) {
    const int kk  = i >> 4;
    const int col = i & 15;
    const int ci  = kk & 31;
    const int kw  = kk >> 5;
    ldsB[(kk >> 1) * 32 + col * 2 + (kk & 1)] =
        w[(size_t)(n0 + col) * KK_ + ci * KW_ + kw];
  }

  // ---- Stage x panel with halo, zero-padding resolved here (branch-free
  //      value selects on clamped addresses; no predication in the K loop).
  for (int i = tid; i < XROWS * 8; i += 256) {       // 8 x float4 per row
    const int row = i >> 3;
    const int c4  = (i & 7) << 2;
    const int t   = t0_block - 1 + row;
    const int tc  = min(max(t, 0), T_ - 1);          // always-valid address
    const float4* p = (const float4*)(xb + tc * CIN_ + c4);
    float4 v = *p;
    const bool ok = (t >= 0) && (t < T_);
    v.x = ok ? v.x : 0.f;  v.y = ok ? v.y : 0.f;
    v.z = ok ? v.z : 0.f;  v.w = ok ? v.w : 0.f;
    *(float4*)(&ldsX[row * CIN_ + c4]) = v;
  }
  __syncthreads();

  const int lo = lane & 15;
  const int hi = lane >> 4;
  const int n  = n0 + lo;                 // this lane's output column
  const int rowBase = (wave << 4) + lo;   // local time row (before +kw)

  v8f acc = {0.f, 0.f, 0.f, 0.f, 0.f, 0.f, 0.f, 0.f};

  // A 16x4 f32 fragment: vgpr j holds K = k0 + 2*hi + j for row M = lane&15.
  // The two elements per lane are (ci, ci+1) in the same tap row -> one b64.
#pragma unroll
  for (int k0 = 0; k0 < KK_; k0 += 4) {
    const int kk0 = k0 + (hi << 1);       // even; pair is {kk0, kk0+1}
    const int kw  = kk0 >> 5;
    const int ci  = kk0 & 31;
    v2f av = *(const v2f*)(&ldsX[(rowBase + kw) * CIN_ + ci]);
    v2f bv = *(const v2f*)(&ldsB[(kk0 >> 1) * 32 + (lo << 1)]);
    acc = __builtin_amdgcn_wmma_f32_16x16x4_f32(
        /*neg_a=*/false, av, /*neg_b=*/false, bv,
        /*c_mod=*/(short)0, acc, /*reuse_a=*/false, /*reuse_b=*/false);
  }

  // ---- Fold bias + BatchNorm(eval): y = conv*sc + sh
  const float sc = gma[n] * rsqrtf(var[n] + 1e-5f);
  const float sh = (cb[n] - mu[n]) * sc + bta[n];
  const int ts   = n >> 7;               // c_out = ts*HID + hid (tile never
  const int hidc = n & (HID_ - 1);       // crosses a 128 boundary)

  // C/D layout: vgpr r -> (M = r + 8*hi, N = lane&15) within the tile.
  const int m0 = m0_block + (wave << 4);
#pragma unroll
  for (int r = 0; r < 8; ++r) {
    const int m  = m0 + r + (hi << 3);
    const int tt = m & (T_ - 1);
    const int s  = tt * TS_ + ts;        // time-major step index
    h[(size_t)s * NCHAIN + b * HID_ + hidc] = acc[r] * sc + sh;
  }
}

// ---------------------------------------------------------------------------
// Kernel 2: LIF scan (snntorch Leaky, reset-by-subtraction, threshold=1).
//   One thread per (b, hid) chain; 2048 sequential steps; reads/writes
//   fully coalesced across the 8192 chains; unroll x8 batches the loads
//   ahead of the dependent FMA chain.
// ---------------------------------------------------------------------------
__global__ __launch_bounds__(256) void lif_scan_kernel(
    const float* __restrict__ h, const float* __restrict__ lif_beta,
    float* __restrict__ spk, float* __restrict__ memo) {
  const int tid = blockIdx.x * 256 + threadIdx.x;  // 0..8191
  const int hid = tid & (HID_ - 1);
  float beta = lif_beta[hid];
  beta = fminf(fmaxf(beta, 0.f), 1.f);
  const float TH = 1.0f;
  float mem = 0.f;
#pragma unroll 8
  for (int s = 0; s < NSTEP; ++s) {
    const float hv    = h[(size_t)s * NCHAIN + tid];
    const float reset = (mem > TH) ? TH : 0.f;       // from previous mem
    mem = __builtin_fmaf(beta, mem, hv) - reset;
    const float sp = (mem - TH > 0.f) ? 1.f : 0.f;
    spk[(size_t)s * NCHAIN + tid]  = sp;
    memo[(size_t)s * NCHAIN + tid] = mem;
  }
}

// ---------------------------------------------------------------------------
extern "C" void kernel_launch(void* const* d_in, const int* in_sizes, int n_in,
                              void* d_out, int out_size, void* d_ws, size_t ws_size,
                              hipStream_t stream) {
  const float* x   = (const float*)d_in[0];  // (B,T,CIN)
  const float* w   = (const float*)d_in[1];  // (COUT,CIN,K)
  const float* cb  = (const float*)d_in[2];  // (COUT,)
  const float* g   = (const float*)d_in[3];  // bn_gamma
  const float* bb  = (const float*)d_in[4];  // bn_beta
  const float* mu  = (const float*)d_in[5];  // bn_mean
  const float* var = (const float*)d_in[6];  // bn_var
  const float* lb  = (const float*)d_in[7];  // lif_beta (HID,)

  float* h   = (float*)d_ws;                          // 2048*8192 f32 = 64 MiB
  float* spk = (float*)d_out;                         // (2048,64,128)
  float* mem = spk + (size_t)NSTEP * NCHAIN;          // (2048,64,128)

  // 8192 blocks: 32 N-tiles x 256 M row-blocks; 8 waves (8 M-tiles) per block.
  conv_bn_wmma_kernel<<<8192, 256, 0, stream>>>(x, w, cb, g, bb, mu, var, h);
  // 8192 chains / 256 threads.
  lif_scan_kernel<<<NCHAIN / 256, 256, 0, stream>>>(h, lb, spk, mem);
}